// ConvpassHist_23630910062603
// MI455X (gfx1250) — compile-verified
//
#include <hip/hip_runtime.h>
#include <hip/hip_bf16.h>

typedef float v2f __attribute__((ext_vector_type(2)));
typedef float v8f __attribute__((ext_vector_type(8)));

#define GROUPS 196
#define TOK    197
#define CDIM   768

__device__ __forceinline__ float qgelu(float v) {
    // x * sigmoid(1.702 x) == x / (1 + exp(-1.702 x))
    return v / (1.0f + __expf(-1.702f * v));
}

// One block per (group g, batch b): fused conv3x3(grouped) + gelu + 4-bin soft
// histogram + 3x3 exclude-pad avgpool + gelu + 12->3 projection.
// Conv and projection run on the matrix pipe as f32 WMMA (16x16x4) GEMMs:
//   conv: A = 3x27 weights (rows padded to 16, K padded to 28),
//         B = 27x16 im2col column tile (one image row per WMMA D-tile)
//   proj: A = 3x12 weights, B = 12x16 pooled activations
__global__ __launch_bounds__(256)
void convpass_fused(const float* __restrict__ x,
                    const float* __restrict__ conv_w,
                    const float* __restrict__ conv_b,
                    const float* __restrict__ centers,
                    const float* __restrict__ widths,
                    const float* __restrict__ down_w,
                    const float* __restrict__ down_b,
                    float* __restrict__ out)
{
    // Channel 3 of xin is a permanently-zero plane: the K=27 pad column of the
    // im2col B-matrix reads from it, so no bounds checks are needed.
    __shared__ float xin[4][18][18];   // zero-padded input tile + zero plane
    __shared__ float wc[16][28];       // conv A-tile: rows 3..15 zero, K 27->28
    __shared__ float wd[16][12];       // proj A-tile: rows 3..15 zero
    __shared__ int   offs[28];         // im2col k -> flat xin offset
    __shared__ float ybuf[3][256];     // conv+gelu output
    __shared__ float hbuf[12][256];    // histogram, later overwritten by pooled
    __shared__ float ctr[12], wid[12], cb[4], db[4];

    const int tid = threadIdx.x;
    const int g   = blockIdx.x;
    const int bb  = blockIdx.y;
    const int h   = tid >> 4, w = tid & 15;

    float* xf  = &xin[0][0][0];
    float* wcf = &wc[0][0];
    float* wdf = &wd[0][0];

    const float* xt = x + (size_t)(bb * TOK + 1 + g) * CDIM;

    // ---- stage 0: zero LDS tiles, then load input tile + params ----
    for (int i = tid; i < 4 * 324; i += 256) xf[i]  = 0.0f;
    for (int i = tid; i < 16 * 28; i += 256) wcf[i] = 0.0f;
    if (tid < 16 * 12) wdf[tid] = 0.0f;
    if (tid < 28) {
        const int k = tid;
        offs[k] = (k < 27) ? (k / 9) * 324 + ((k % 9) / 3) * 18 + (k % 3)
                           : 3 * 324;                // pad column -> zero plane
    }
    __syncthreads();

    #pragma unroll
    for (int c = 0; c < 3; ++c)
        xin[c][h + 1][w + 1] = xt[c * 256 + tid];    // coalesced 1KB load per channel
    if (tid < 81) wcf[(tid / 27) * 28 + (tid % 27)] = conv_w[g * 81 + tid];
    if (tid < 36) wdf[(tid / 12) * 12 + (tid % 12)] = down_w[g * 36 + tid];
    if (tid < 12) { ctr[tid] = centers[g * 12 + tid]; wid[tid] = widths[g * 12 + tid]; }
    if (tid < 3)  { cb[tid] = conv_b[g * 3 + tid];   db[tid] = down_b[g * 3 + tid]; }
    __syncthreads();

    const int lane = tid & 31;
    const int wave = tid >> 5;
    const int lh   = lane & 15;          // B column / A row-within-half
    const int khi  = (lane >> 4) << 1;   // lanes 0-15: K+0/K+1 ; 16-31: K+2/K+3

    // ---- stage 1: conv as im2col GEMM (3x27 @ 27x16) via f32 WMMA ----
    // Hoist row-invariant per-lane operands: A values and B im2col offsets.
    float a0[7], a1[7];
    int   bo0[7], bo1[7];
    #pragma unroll
    for (int j = 0; j < 7; ++j) {
        const int k0 = 4 * j + khi;
        a0[j]  = wc[lh][k0];
        a1[j]  = wc[lh][k0 + 1];
        bo0[j] = offs[k0];
        bo1[j] = offs[k0 + 1];
    }

    #pragma unroll
    for (int r = 0; r < 2; ++r) {
        const int hr  = wave * 2 + r;            // image row of this D-tile
        const int col = hr * 18 + lh;            // per-lane im2col base
        v8f acc = {0.f, 0.f, 0.f, 0.f, 0.f, 0.f, 0.f, 0.f};
        #pragma unroll
        for (int j = 0; j < 7; ++j) {
            v2f av, bv;
            av.x = a0[j];
            av.y = a1[j];
            bv.x = xf[bo0[j] + col];
            bv.y = xf[bo1[j] + col];
            acc = __builtin_amdgcn_wmma_f32_16x16x4_f32(false, av, false, bv,
                                                        (short)0, acc, false, false);
        }
        if (lane < 16) {             // D rows 0..2 live in VGPRs 0..2, lanes 0-15
            #pragma unroll
            for (int o = 0; o < 3; ++o)
                ybuf[o][hr * 16 + lane] = qgelu(acc[o] + cb[o]);
        }
    }
    __syncthreads();

    // ---- stage 2: 4-bin soft histogram + normalize (per pixel) ----
    #pragma unroll
    for (int o = 0; o < 3; ++o) {
        const float yv = ybuf[o][tid];
        float v[4]; float s = 0.f;
        #pragma unroll
        for (int k = 0; k < 4; ++k) {
            const float t = fabsf(yv + ctr[o * 4 + k]);
            const float u = fmaxf(fmaf(wid[o * 4 + k], t, 1.0f), 0.0f);
            v[k] = u; s += u;
        }
        const float inv = 1.0f / (s + 1e-5f);
        #pragma unroll
        for (int k = 0; k < 4; ++k)
            hbuf[o * 4 + k][tid] = v[k] * inv;
    }
    __syncthreads();

    // ---- stage 3: 3x3 avgpool (exclude pad) + gelu, in registers ----
    float pg[12];
    {
        const int h0 = (h > 0) ? h - 1 : 0, h1 = (h < 15) ? h + 1 : 15;
        const int w0 = (w > 0) ? w - 1 : 0, w1 = (w < 15) ? w + 1 : 15;
        const float rc = 1.0f / (float)((h1 - h0 + 1) * (w1 - w0 + 1));
        #pragma unroll
        for (int i = 0; i < 12; ++i) {
            float s = 0.f;
            for (int hh = h0; hh <= h1; ++hh)
                for (int ww = w0; ww <= w1; ++ww)
                    s += hbuf[i][hh * 16 + ww];
            pg[i] = qgelu(s * rc);
        }
    }
    __syncthreads();
    #pragma unroll
    for (int i = 0; i < 12; ++i) hbuf[i][tid] = pg[i];   // reuse hbuf
    __syncthreads();

    // ---- stage 4: 12->3 projection (3x12 @ 12x16) via f32 WMMA ----
    float da0[3], da1[3];
    #pragma unroll
    for (int j = 0; j < 3; ++j) {
        const int k0 = 4 * j + khi;
        da0[j] = wd[lh][k0];
        da1[j] = wd[lh][k0 + 1];
    }

    float* op = out + (size_t)(bb * TOK + 1 + g) * CDIM;
    #pragma unroll
    for (int r = 0; r < 2; ++r) {
        const int hr = wave * 2 + r;
        v8f acc = {0.f, 0.f, 0.f, 0.f, 0.f, 0.f, 0.f, 0.f};
        #pragma unroll
        for (int j = 0; j < 3; ++j) {
            const int k0 = 4 * j + khi;
            v2f av, bv;
            av.x = da0[j];
            av.y = da1[j];
            bv.x = hbuf[k0][hr * 16 + lh];
            bv.y = hbuf[k0 + 1][hr * 16 + lh];
            acc = __builtin_amdgcn_wmma_f32_16x16x4_f32(false, av, false, bv,
                                                        (short)0, acc, false, false);
        }
        if (lane < 16) {
            #pragma unroll
            for (int o = 0; o < 3; ++o)
                op[o * 256 + hr * 16 + lane] = acc[o] + db[o];
        }
    }
}

// CLS token pass-through: out[b,0,:] = x[b,0,:]
__global__ void cls_copy(const float* __restrict__ x, float* __restrict__ out, int n)
{
    const int i = blockIdx.x * blockDim.x + threadIdx.x;
    if (i < n) {
        const int b = i / CDIM, c = i % CDIM;
        out[(size_t)b * TOK * CDIM + c] = x[(size_t)b * TOK * CDIM + c];
    }
}

extern "C" void kernel_launch(void* const* d_in, const int* in_sizes, int n_in,
                              void* d_out, int out_size, void* d_ws, size_t ws_size,
                              hipStream_t stream)
{
    const float* x       = (const float*)d_in[0];
    const float* conv_w  = (const float*)d_in[1];
    const float* conv_b  = (const float*)d_in[2];
    const float* centers = (const float*)d_in[3];
    const float* widths  = (const float*)d_in[4];
    const float* down_w  = (const float*)d_in[5];
    const float* down_b  = (const float*)d_in[6];
    float* out = (float*)d_out;

    const int B = in_sizes[0] / (TOK * CDIM);

    dim3 grid(GROUPS, B);
    convpass_fused<<<grid, 256, 0, stream>>>(x, conv_w, conv_b, centers, widths,
                                             down_w, down_b, out);

    const int n = B * CDIM;
    cls_copy<<<(n + 255) / 256, 256, 0, stream>>>(x, out, n);
}